// AndersonDarlingDistance_4552665334327
// MI455X (gfx1250) — compile-verified
//
#include <hip/hip_runtime.h>
#include <hip/hip_bf16.h>

// Anderson-Darling distance, N=65536 rows x D=256 dims.
// Pipeline:
//   K1: per-column LDS-resident bitonic sort (256 KB column fits in 320 KB WGP LDS)
//       for both X and X_hat -> workspace (contiguous per-column layout).
//   K2: per column: stage sorted X column into LDS via global_load_async_to_lds,
//       branchless 17-probe binary search per element (2-way interleaved for ILP),
//       term = (2i+1)*log(u) + (2N-1-2i)*log(1-u); WMMA f32 16x16x4 with A=ones
//       does the cross-lane partial reduction; per-column sum -> block_sums[col].
//   K3: reduce 256 column sums, out = -total/(N*D) - N.

#define N_ROWS 65536
#define N_DIMS 256
#define SORT_THREADS 1024

typedef __attribute__((ext_vector_type(2))) float v2f;
typedef __attribute__((ext_vector_type(8))) float v8f;

// Exact parameter types for __builtin_amdgcn_global_load_async_to_lds_b128:
// int4 (vector_size 16) pointers, global (AS1) source and LDS (AS3) dest.
typedef int v4i __attribute__((vector_size(16)));
typedef __attribute__((address_space(1))) v4i g_v4i;
typedef __attribute__((address_space(3))) v4i l_v4i;

#if defined(__has_builtin)
#if __has_builtin(__builtin_amdgcn_global_load_async_to_lds_b128)
#define HAVE_ASYNC_LDS 1
#endif
#if __has_builtin(__builtin_amdgcn_s_wait_asynccnt)
#define HAVE_WAIT_ASYNC 1
#endif
#endif

// ---------------- Kernel 1: bitonic sort of one column in LDS ----------------
__global__ __launch_bounds__(SORT_THREADS)
void ad_sort_col(const float* __restrict__ X, const float* __restrict__ Xh,
                 float* __restrict__ xs, float* __restrict__ xh) {
  extern __shared__ float s[];
  const unsigned tid = threadIdx.x;
  const unsigned T   = blockDim.x;
  unsigned col = blockIdx.x;
  const float* src;
  float* dst;
  if (col < N_DIMS) {
    src = X;
    dst = xs + (size_t)col * N_ROWS;
  } else {
    col -= N_DIMS;
    src = Xh;
    dst = xh + (size_t)col * N_ROWS;
  }
  // Strided column gather (L2-absorbed: X fits in 192MB L2).
  for (unsigned i = tid; i < N_ROWS; i += T)
    s[i] = src[(size_t)i * N_DIMS + col];
  __syncthreads();

  // Bitonic sort, ascending. 136 passes over 32768 pairs.
  for (unsigned k = 2; k <= N_ROWS; k <<= 1) {
    for (unsigned j = k >> 1; j > 0; j >>= 1) {
      for (unsigned t = tid; t < (N_ROWS / 2); t += T) {
        unsigned i = t + (t & ~(j - 1));  // (t/j)*2j + t%j
        unsigned p = i | j;
        float a = s[i];
        float b = s[p];
        bool up = ((i & k) == 0);
        if ((a > b) == up) { s[i] = b; s[p] = a; }
      }
      __syncthreads();
    }
  }
  // Coalesced contiguous store of the sorted column.
  for (unsigned i = tid; i < N_ROWS; i += T)
    dst[i] = s[i];
}

// Branchless count of {s[0..N-1] <= v}: 16 in-range probes + 1 fixup probe.
// Before processing step S, pos <= 65536 - 2S, so probe index pos+S-1 <= 65535-S.
__device__ __forceinline__ unsigned lds_count_le(const float* s, float v) {
  unsigned pos = 0;
#pragma unroll
  for (unsigned step = N_ROWS / 2; step >= 1; step >>= 1)
    pos += (s[pos + step - 1] <= v) ? step : 0u;
  pos += (s[pos] <= v) ? 1u : 0u;  // pos <= 65535 here, always in range
  return pos;                      // in [0, N]
}

// ---- Kernel 2: counts via LDS binary search + log terms + WMMA reduction ----
__global__ __launch_bounds__(SORT_THREADS)
void ad_count_reduce(const float* __restrict__ xs, const float* __restrict__ xh,
                     double* __restrict__ block_sums) {
  extern __shared__ float s[];
  const unsigned tid = threadIdx.x;
  const unsigned T   = blockDim.x;
  const unsigned col = blockIdx.x;
  const float* xcol = xs + (size_t)col * N_ROWS;
  const float* hcol = xh + (size_t)col * N_ROWS;

  // Stage sorted X column (256 KB) into LDS.
#if defined(HAVE_ASYNC_LDS)
  for (unsigned i = tid * 4; i < N_ROWS; i += T * 4) {
    __builtin_amdgcn_global_load_async_to_lds_b128(
        (g_v4i*)(xcol + i), (l_v4i*)(s + i), 0, 0);
  }
#if defined(HAVE_WAIT_ASYNC)
  __builtin_amdgcn_s_wait_asynccnt(0);
#else
  asm volatile("s_wait_asynccnt 0" ::: "memory");
#endif
#else
  for (unsigned i = tid; i < N_ROWS; i += T)
    s[i] = xcol[i];
#endif
  __syncthreads();

  const float inv = 1.0f / (float)(N_ROWS + 2);
  float local = 0.0f;
  // Two independent searches per iteration: overlapping ds-load dependency
  // chains hides LDS latency (each search is 17 serially dependent probes).
  for (unsigned i0 = tid; i0 < N_ROWS; i0 += 2 * T) {
    const unsigned i1 = i0 + T;  // N_ROWS / T = 64 is even -> always valid
    const float v0 = hcol[i0];
    const float v1 = hcol[i1];
    const unsigned c0 = lds_count_le(s, v0);
    const unsigned c1 = lds_count_le(s, v1);

    const float u0  = (float)(c0 + 1) * inv;            // (cnt+1)/(N+2)
    const float um0 = (float)(N_ROWS + 1 - c0) * inv;   // 1 - u, exactly
    const float w0  = 2.0f * (float)i0 + 1.0f;          // coef (2i+1)
    local += w0 * __logf(u0) + (2.0f * (float)N_ROWS - w0) * __logf(um0);

    const float u1  = (float)(c1 + 1) * inv;
    const float um1 = (float)(N_ROWS + 1 - c1) * inv;
    const float w1  = 2.0f * (float)i1 + 1.0f;
    local += w1 * __logf(u1) + (2.0f * (float)N_ROWS - w1) * __logf(um1);
  }
  __syncthreads();  // done reading sorted data from LDS

  // Cross-lane partial reduction via WMMA: A = ones(16x4), B carries the 32
  // per-lane partials (slot 1 zero), C = 0. Every row of D holds the 16
  // column-sums of B, so summing c[0] over lanes 0..15 = wave total. This is
  // exact for any B lane layout because A is uniform.
  v2f a; a.x = 1.0f; a.y = 1.0f;
  v2f b; b.x = local; b.y = 0.0f;
  v8f c = {};
  c = __builtin_amdgcn_wmma_f32_16x16x4_f32(
      /*neg_a=*/false, a, /*neg_b=*/false, b,
      /*c_mod=*/(short)0, c, /*reuse_a=*/false, /*reuse_b=*/false);

  const unsigned lane = tid & 31u;
  const unsigned wave = tid >> 5;
  float* red = s;  // reuse LDS
  red[wave * 32 + lane] = c[0];
  __syncthreads();
  if (tid == 0) {
    double sum = 0.0;
    const unsigned nwaves = T >> 5;
    for (unsigned w = 0; w < nwaves; ++w)
      for (unsigned l = 0; l < 16; ++l)   // lanes 0..15 hold D row 0
        sum += (double)red[w * 32 + l];
    block_sums[col] = sum;                // deterministic: fixed slot per column
  }
}

// ------------------- Kernel 3: final reduction over columns ------------------
__global__ __launch_bounds__(N_DIMS)
void ad_finalize(const double* __restrict__ block_sums, float* __restrict__ out) {
  __shared__ double red[N_DIMS];
  const unsigned tid = threadIdx.x;
  red[tid] = block_sums[tid];
  __syncthreads();
  for (unsigned off = N_DIMS / 2; off > 0; off >>= 1) {
    if (tid < off) red[tid] += red[tid + off];
    __syncthreads();
  }
  if (tid == 0) {
    const double total = red[0];  // sum_d S_d, with w_d = -S_d / N
    const double mean_w = -total / ((double)N_ROWS * (double)N_DIMS);
    out[0] = (float)(mean_w - (double)N_ROWS);
  }
}

extern "C" void kernel_launch(void* const* d_in, const int* in_sizes, int n_in,
                              void* d_out, int out_size, void* d_ws, size_t ws_size,
                              hipStream_t stream) {
  const float* X  = (const float*)d_in[0];
  const float* Xh = (const float*)d_in[1];
  float* out = (float*)d_out;

  char* ws = (char*)d_ws;
  double* block_sums = (double*)ws;                       // 2 KB
  float* xs = (float*)(ws + 4096);                        // 64 MB sorted X
  float* xh = xs + (size_t)N_ROWS * N_DIMS;               // 64 MB sorted X_hat

  const size_t lds_bytes = (size_t)N_ROWS * sizeof(float);  // 256 KB / WG

  ad_sort_col<<<2 * N_DIMS, SORT_THREADS, lds_bytes, stream>>>(X, Xh, xs, xh);
  ad_count_reduce<<<N_DIMS, SORT_THREADS, lds_bytes, stream>>>(xs, xh, block_sums);
  ad_finalize<<<1, N_DIMS, 0, stream>>>(block_sums, out);
}